// GNNPolicy_3384434230027
// MI455X (gfx1250) — compile-verified
//
#include <hip/hip_runtime.h>
#include <hip/hip_bf16.h>

typedef __attribute__((ext_vector_type(2))) float v2f;
typedef __attribute__((ext_vector_type(8))) float v8f;

#define HID 128
#define NGRAPH 64

// ---------------- small utility kernels ----------------

__global__ void k_fill(float* p, int n, float v) {
    int i = blockIdx.x * 256 + threadIdx.x;
    if (i < n) p[i] = v;
}

__global__ void k_deg_count(const int* __restrict__ dst, int E, float* __restrict__ deg) {
    int e = blockIdx.x * 256 + threadIdx.x;
    if (e < E) atomicAdd(deg + dst[e], 1.0f);
}

__global__ void k_rsqrt(float* d, int n) {
    int i = blockIdx.x * 256 + threadIdx.x;
    if (i < n) d[i] = rsqrtf(d[i]);
}

// layer-1 X@W fused as embedding gather: xw = W[res] + pos0*W[20] + pos1*W[21] + pos2*W[22]
// then y = xw * dinv[row]
__global__ void k_embed(const int* __restrict__ res, const float* __restrict__ pos,
                        const float* __restrict__ W, const float* __restrict__ dinv,
                        int N, float* __restrict__ y) {
    int idx = blockIdx.x * 256 + threadIdx.x;
    if (idx >= N * HID) return;
    int i = idx >> 7, f = idx & 127;
    float xw = W[res[i] * HID + f]
             + pos[i * 3 + 0] * W[20 * HID + f]
             + pos[i * 3 + 1] * W[21 * HID + f]
             + pos[i * 3 + 2] * W[22 * HID + f];
    y[idx] = xw * dinv[i];
}

// one wave (32 lanes) per edge; each lane moves 4 contiguous floats
__global__ void k_scatter(const int* __restrict__ src, const int* __restrict__ dst,
                          int E, const float* __restrict__ y, float* __restrict__ s) {
    int e = blockIdx.x * 8 + (threadIdx.x >> 5);
    if (e >= E) return;
    int lane = threadIdx.x & 31;
    size_t so = (size_t)src[e] * HID + lane * 4;
    size_t doo = (size_t)dst[e] * HID + lane * 4;
    float4 v = *(const float4*)(y + so);
    float* p = s + doo;
    atomicAdd(p + 0, v.x);
    atomicAdd(p + 1, v.y);
    atomicAdd(p + 2, v.z);
    atomicAdd(p + 3, v.w);
}

// h = relu(dinv[i] * (s + y) + b[f])
__global__ void k_combine(const float* __restrict__ y, const float* __restrict__ s,
                          const float* __restrict__ dinv, const float* __restrict__ b,
                          int N, float* __restrict__ h) {
    int idx = blockIdx.x * 256 + threadIdx.x;
    if (idx >= N * HID) return;
    int i = idx >> 7, f = idx & 127;
    h[idx] = fmaxf(dinv[i] * (s[idx] + y[idx]) + b[f], 0.0f);
}

__global__ void k_cnt(const int* __restrict__ batch, int N, float* __restrict__ cnt) {
    int i = blockIdx.x * 256 + threadIdx.x;
    if (i < N) atomicAdd(cnt + batch[i], 1.0f);
}

__global__ void k_pool_sum(const float* __restrict__ h, const int* __restrict__ batch,
                           int N, float* __restrict__ pesum) {
    int idx = blockIdx.x * 256 + threadIdx.x;
    if (idx >= N * HID) return;
    int i = idx >> 7, f = idx & 127;
    atomicAdd(pesum + batch[i] * HID + f, h[idx]);
}

__global__ void k_pool_div(const float* __restrict__ pesum, const float* __restrict__ cnt,
                           float* __restrict__ pe) {
    int idx = blockIdx.x * 256 + threadIdx.x;
    if (idx >= NGRAPH * HID) return;
    pe[idx] = pesum[idx] / fmaxf(cnt[idx >> 7], 1.0f);
}

// feat = [ h2_m | pe[batch_m] ]  -> [N_M, 256]
__global__ void k_feat(const float* __restrict__ h, const float* __restrict__ pe,
                       const int* __restrict__ batch, int N, float* __restrict__ feat) {
    int idx = blockIdx.x * 256 + threadIdx.x;
    if (idx >= N * 2 * HID) return;
    int i = idx >> 8, f = idx & 255;
    feat[idx] = (f < HID) ? h[(size_t)i * HID + f]
                          : pe[batch[i] * HID + (f - HID)];
}

// out[i,:] = hmid[i,:] @ W2[128,3] + b2
__global__ void k_head(const float* __restrict__ hmid, const float* __restrict__ W2,
                       const float* __restrict__ b2, int N, float* __restrict__ out) {
    int i = blockIdx.x * 256 + threadIdx.x;
    if (i >= N) return;
    float a0 = b2[0], a1 = b2[1], a2 = b2[2];
    const float* h = hmid + (size_t)i * HID;
    for (int k = 0; k < HID; ++k) {
        float hv = h[k];
        a0 += hv * W2[k * 3 + 0];
        a1 += hv * W2[k * 3 + 1];
        a2 += hv * W2[k * 3 + 2];
    }
    out[(size_t)i * 3 + 0] = a0;
    out[(size_t)i * 3 + 1] = a1;
    out[(size_t)i * 3 + 2] = a2;
}

// ---------------- W -> WMMA-fragment-order swizzle ----------------
// Wf layout: for k-step kk (4 K-rows), lane l holds 16 contiguous floats:
//   Wf[(kk*32 + l)*16 + nt*2 + j] = W[(4*kk + (l>>4)*2 + j)*128 + nt*16 + (l&15)]
// so each lane fetches all 8 B-fragments of a k-step with 4 b128 loads.
__global__ void k_wfrag(const float* __restrict__ W, int K, float* __restrict__ Wf) {
    int idx = blockIdx.x * 256 + threadIdx.x;
    if (idx >= K * HID) return;
    int j    = idx & 1;
    int nt   = (idx >> 1) & 7;
    int l    = (idx >> 4) & 31;
    int kk   = idx >> 9;
    int krow = 4 * kk + (l >> 4) * 2 + j;
    int col  = nt * 16 + (l & 15);
    Wf[idx] = W[(size_t)krow * HID + col];
}

// ---------------- WMMA f32 GEMM: out[M,128] = A[M,K] @ Wf(K,128 swizzled) ----------------
// Register-blocked: single pass over K, 8 live v8f accumulators.
// Per k-step: 1 b64 A-frag load + 4 b128 fragment loads -> 8 chained WMMAs.
// Per V_WMMA_F32_16X16X4_F32 VGPR layout:
//   A frag (16x4):  lane l (l&15 = M row), koff = (l>>4)*2, vgprs = {K+koff, K+koff+1}
//   B frag (4x16):  lane l (l&15 = N col), vgprs = rows {K+koff, K+koff+1} of W
//   C/D (16x16):    vgpr r -> row r (lanes 0-15) / row r+8 (lanes 16-31)
__global__ void k_gemm128(const float* __restrict__ A, int K, int M,
                          const float* __restrict__ Wf,
                          const float* __restrict__ rowscale,
                          const float* __restrict__ bias, int relu,
                          float* __restrict__ out) {
    int w = threadIdx.x >> 5;
    int lane = threadIdx.x & 31;
    int tile = blockIdx.x * 4 + w;
    if (tile * 16 >= M) return;          // whole wave exits together; EXEC stays all-1s
    int row0 = tile * 16;
    int half = lane >> 4;                 // 0 or 1
    int lmod = lane & 15;
    int koff = half * 2;
    const float* Arow = A + (size_t)(row0 + lmod) * K + koff;
    const float* Wl   = Wf + (size_t)lane * 16;

    v8f acc[8];
    #pragma unroll
    for (int nt = 0; nt < 8; ++nt) acc[nt] = (v8f){};

    int ksteps = K >> 2;
    for (int kk = 0; kk < ksteps; ++kk) {
        v2f a;
        a.x = Arow[4 * kk];
        a.y = Arow[4 * kk + 1];
        const float4* q = (const float4*)(Wl + (size_t)kk * 512);
        float4 q0 = q[0], q1 = q[1], q2 = q[2], q3 = q[3];
        v2f b0; b0.x = q0.x; b0.y = q0.y;
        v2f b1; b1.x = q0.z; b1.y = q0.w;
        v2f b2; b2.x = q1.x; b2.y = q1.y;
        v2f b3; b3.x = q1.z; b3.y = q1.w;
        v2f b4; b4.x = q2.x; b4.y = q2.y;
        v2f b5; b5.x = q2.z; b5.y = q2.w;
        v2f b6; b6.x = q3.x; b6.y = q3.y;
        v2f b7; b7.x = q3.z; b7.y = q3.w;
        acc[0] = __builtin_amdgcn_wmma_f32_16x16x4_f32(false, a, false, b0, (short)0, acc[0], false, false);
        acc[1] = __builtin_amdgcn_wmma_f32_16x16x4_f32(false, a, false, b1, (short)0, acc[1], false, false);
        acc[2] = __builtin_amdgcn_wmma_f32_16x16x4_f32(false, a, false, b2, (short)0, acc[2], false, false);
        acc[3] = __builtin_amdgcn_wmma_f32_16x16x4_f32(false, a, false, b3, (short)0, acc[3], false, false);
        acc[4] = __builtin_amdgcn_wmma_f32_16x16x4_f32(false, a, false, b4, (short)0, acc[4], false, false);
        acc[5] = __builtin_amdgcn_wmma_f32_16x16x4_f32(false, a, false, b5, (short)0, acc[5], false, false);
        acc[6] = __builtin_amdgcn_wmma_f32_16x16x4_f32(false, a, false, b6, (short)0, acc[6], false, false);
        acc[7] = __builtin_amdgcn_wmma_f32_16x16x4_f32(false, a, false, b7, (short)0, acc[7], false, false);
    }

    float rs[8];
    if (rowscale) {
        #pragma unroll
        for (int r = 0; r < 8; ++r) rs[r] = rowscale[row0 + r + half * 8];
    }

    #pragma unroll
    for (int nt = 0; nt < 8; ++nt) {
        int col = nt * 16 + lmod;
        float bv = bias ? bias[col] : 0.0f;
        #pragma unroll
        for (int r = 0; r < 8; ++r) {
            float v = acc[nt][r];
            if (rowscale) v *= rs[r];
            v += bv;
            if (relu) v = fmaxf(v, 0.0f);
            out[(size_t)(row0 + r + half * 8) * HID + col] = v;
        }
    }
}

// ---------------- host orchestration ----------------

static inline int cdiv(long long a, long long b) { return (int)((a + b - 1) / b); }

static void run_branch(const int* res, const float* pos, const int* ei,
                       int N, int E,
                       const float* W1c, const float* b1c,
                       const float* W2c, const float* b2c,
                       float* dinv, float* A, float* B, float* C, float* Wf,
                       hipStream_t stream) {
    const int* src = ei;
    const int* dst = ei + E;
    size_t nh = (size_t)N * HID;

    k_fill<<<cdiv(N, 256), 256, 0, stream>>>(dinv, N, 1.0f);
    k_deg_count<<<cdiv(E, 256), 256, 0, stream>>>(dst, E, dinv);
    k_rsqrt<<<cdiv(N, 256), 256, 0, stream>>>(dinv, N);

    // layer 1
    k_embed<<<cdiv(nh, 256), 256, 0, stream>>>(res, pos, W1c, dinv, N, A);
    hipMemsetAsync(B, 0, nh * sizeof(float), stream);
    k_scatter<<<cdiv(E, 8), 256, 0, stream>>>(src, dst, E, A, B);
    k_combine<<<cdiv(nh, 256), 256, 0, stream>>>(A, B, dinv, b1c, N, C);

    // layer 2: y2 = (h1 @ W2) * dinv  (WMMA), then scatter + combine
    k_wfrag<<<cdiv(HID * HID, 256), 256, 0, stream>>>(W2c, HID, Wf);
    k_gemm128<<<cdiv(N / 16, 4), 128, 0, stream>>>(C, HID, N, Wf, dinv, nullptr, 0, A);
    hipMemsetAsync(B, 0, nh * sizeof(float), stream);
    k_scatter<<<cdiv(E, 8), 256, 0, stream>>>(src, dst, E, A, B);
    k_combine<<<cdiv(nh, 256), 256, 0, stream>>>(A, B, dinv, b2c, N, C);
}

extern "C" void kernel_launch(void* const* d_in, const int* in_sizes, int n_in,
                              void* d_out, int out_size, void* d_ws, size_t ws_size,
                              hipStream_t stream) {
    (void)n_in; (void)out_size; (void)ws_size;

    const int*   res_p   = (const int*)d_in[0];
    const float* pos_p   = (const float*)d_in[1];
    const int*   ei_p    = (const int*)d_in[2];
    const int*   batch_p = (const int*)d_in[3];
    const int*   res_m   = (const int*)d_in[4];
    const float* pos_m   = (const float*)d_in[5];
    const int*   ei_m    = (const int*)d_in[6];
    const int*   batch_m = (const int*)d_in[7];
    const float* W_p1 = (const float*)d_in[8];
    const float* b_p1 = (const float*)d_in[9];
    const float* W_p2 = (const float*)d_in[10];
    const float* b_p2 = (const float*)d_in[11];
    const float* W_m1 = (const float*)d_in[12];
    const float* b_m1 = (const float*)d_in[13];
    const float* W_m2 = (const float*)d_in[14];
    const float* b_m2 = (const float*)d_in[15];
    const float* W1f  = (const float*)d_in[16];
    const float* b1f  = (const float*)d_in[17];
    const float* W2f  = (const float*)d_in[18];
    const float* b2f  = (const float*)d_in[19];

    int N_P = in_sizes[0];
    int E_P = in_sizes[2] / 2;
    int N_M = in_sizes[4];
    int E_M = in_sizes[6] / 2;

    // workspace layout (floats); A/B/C sized for the protein branch, reused by
    // the molecule branch (smaller), feat aliases A, hmid aliases B.
    float* ws = (float*)d_ws;
    size_t nhp = (size_t)N_P * HID;
    float* A      = ws;
    float* B      = A + nhp;
    float* C      = B + nhp;
    float* dinv_p = C + nhp;
    float* dinv_m = dinv_p + N_P;
    float* pe_sum = dinv_m + N_M;
    float* pe     = pe_sum + NGRAPH * HID;
    float* cnt    = pe + NGRAPH * HID;
    float* Wfrag  = cnt + NGRAPH;          // up to 256*128 floats (128 KB)

    // ---- protein branch: two GCN layers then mean-pool -> pe[64,128] ----
    run_branch(res_p, pos_p, ei_p, N_P, E_P, W_p1, b_p1, W_p2, b_p2,
               dinv_p, A, B, C, Wfrag, stream);
    hipMemsetAsync(pe_sum, 0, (size_t)NGRAPH * HID * sizeof(float), stream);
    hipMemsetAsync(cnt, 0, NGRAPH * sizeof(float), stream);
    k_cnt<<<cdiv(N_P, 256), 256, 0, stream>>>(batch_p, N_P, cnt);
    k_pool_sum<<<cdiv(nhp, 256), 256, 0, stream>>>(C, batch_p, N_P, pe_sum);
    k_pool_div<<<cdiv(NGRAPH * HID, 256), 256, 0, stream>>>(pe_sum, cnt, pe);

    // ---- molecule branch (reuses A/B/C) -> h2_m in C ----
    run_branch(res_m, pos_m, ei_m, N_M, E_M, W_m1, b_m1, W_m2, b_m2,
               dinv_m, A, B, C, Wfrag, stream);

    // ---- head: feat=[h2_m | pe[batch_m]], hmid=relu(feat@W1f+b1f), out=hmid@W2f+b2f ----
    size_t nfm = (size_t)N_M * 2 * HID;
    float* feat = A;   // N_M*256 floats <= N_P*128 floats
    float* hmid = B;   // N_M*128 floats
    k_feat<<<cdiv(nfm, 256), 256, 0, stream>>>(C, pe, batch_m, N_M, feat);
    k_wfrag<<<cdiv(2 * HID * HID, 256), 256, 0, stream>>>(W1f, 2 * HID, Wfrag);
    k_gemm128<<<cdiv(N_M / 16, 4), 128, 0, stream>>>(feat, 2 * HID, N_M, Wfrag,
                                                     nullptr, b1f, 1, hmid);
    k_head<<<cdiv(N_M, 256), 256, 0, stream>>>(hmid, W2f, b2f, N_M, (float*)d_out);
}